// TFBlock_20718922236173
// MI455X (gfx1250) — compile-verified
//
#include <hip/hip_runtime.h>
#include <cstddef>
#include <cstdint>

typedef float v2f __attribute__((ext_vector_type(2)));
typedef float v8f __attribute__((ext_vector_type(8)));

#define BATCH 8
#define NPIX  4096          // 64*64
#define QKVC  144           // 64 q + 16 k + 64 v channels

// ---------------- workspace layout (floats) ----------------
static constexpr size_t OFF_QKV  = 0;
static constexpr size_t OFF_LAMC = OFF_QKV  + (size_t)BATCH*QKVC*NPIX;
static constexpr size_t OFF_SBS  = OFF_LAMC + (size_t)BATCH*16*64;
static constexpr size_t OFF_SBB  = OFF_SBS  + 144;
static constexpr size_t OFF_H1   = OFF_SBB  + 144;
static constexpr size_t OFF_H2   = OFF_H1   + (size_t)BATCH*64*NPIX;

__device__ __forceinline__ void wait_asynccnt0() {
#if __has_builtin(__builtin_amdgcn_s_wait_asynccnt)
  __builtin_amdgcn_s_wait_asynccnt(0);
#else
  asm volatile("s_wait_asynccnt 0x0" ::: "memory");
#endif
}

// ---------------------------------------------------------------------------
// Fold inference BatchNorm into per-channel scale/bias for the 144 qkv chans.
// ---------------------------------------------------------------------------
__global__ void prep_scalebias(const float* __restrict__ gq, const float* __restrict__ bq,
                               const float* __restrict__ mq, const float* __restrict__ vq,
                               const float* __restrict__ gv, const float* __restrict__ bv,
                               const float* __restrict__ mv, const float* __restrict__ vv,
                               float* __restrict__ s_arr, float* __restrict__ b_arr) {
  int i = threadIdx.x;
  if (i >= 144) return;
  float s = 1.f, bb = 0.f;
  if (i < 64) {                       // q channels
    float sc = gq[i] / sqrtf(vq[i] + 1e-5f);
    s = sc; bb = bq[i] - mq[i] * sc;
  } else if (i >= 80) {               // v channels
    int j = i - 80;
    float sc = gv[j] / sqrtf(vv[j] + 1e-5f);
    s = sc; bb = bv[j] - mv[j] * sc;
  }                                   // k channels: identity
  s_arr[i] = s; b_arr[i] = bb;
}

// ---------------------------------------------------------------------------
// f32 WMMA GEMM:  out[b, outOff+m, n] = W[m,:] . X[b,:,n]  (+scale/bias/add)
// Exact f32 via V_WMMA_F32_16X16X4_F32.  Block = 8 waves; each wave owns a
// 16(M) x 64(N) stripe = 4 WMMA tiles, so one A load feeds 4 WMMAs per k-step.
// A 16x4 layout : lanes 0-15 M=lane (K0,K1), lanes 16-31 M=lane-16 (K2,K3)
// B 4x16 layout : lanes 0-15 N=lane (K0,K1), lanes 16-31 N=lane-16 (K2,K3)
// C/D 16x16     : VGPR r -> M = r + 8*(lane>>4), N = n0 + (lane&15)
// ---------------------------------------------------------------------------
__global__ void gemm_wmma_f32(const float* __restrict__ Wt, const float* __restrict__ X,
                              const float* __restrict__ scale, const float* __restrict__ bias,
                              const float* __restrict__ addp, float* __restrict__ out,
                              int M, int K, int outStride, int outOff) {
  const int lane  = threadIdx.x & 31;
  const int wave  = threadIdx.x >> 5;
  const int batch = blockIdx.z;
  const int m0    = blockIdx.y << 4;
  const int nbase = (blockIdx.x * 8 + wave) * 64;    // 4 consecutive 16-wide N tiles
  const int half  = lane >> 4;     // which K pair this lane supplies
  const int l16   = lane & 15;
  const float* Xb   = X + (size_t)batch * K * NPIX;
  const float* wrow = Wt + (size_t)(m0 + l16) * K + 2 * half;  // 8B aligned (K mult of 4)
  const float* xcol = Xb + nbase + l16;
  __builtin_prefetch(wrow, 0, 1);                    // global_prefetch_b8 on the A row

  v8f c0 = {}, c1 = {}, c2 = {}, c3 = {};
  for (int k0 = 0; k0 < K; k0 += 4) {
    v2f a = *(const v2f*)(wrow + k0);                // W[m][ka], W[m][ka+1]
    const float* xk = xcol + (size_t)(k0 + 2 * half) * NPIX;
    v2f b0 = { xk[0],  xk[NPIX +  0] };
    v2f b1 = { xk[16], xk[NPIX + 16] };
    v2f b2 = { xk[32], xk[NPIX + 32] };
    v2f b3 = { xk[48], xk[NPIX + 48] };
    c0 = __builtin_amdgcn_wmma_f32_16x16x4_f32(false, a, false, b0, (short)0, c0, false, false);
    c1 = __builtin_amdgcn_wmma_f32_16x16x4_f32(false, a, false, b1, (short)0, c1, false, false);
    c2 = __builtin_amdgcn_wmma_f32_16x16x4_f32(false, a, false, b2, (short)0, c2, false, false);
    c3 = __builtin_amdgcn_wmma_f32_16x16x4_f32(false, a, false, b3, (short)0, c3, false, false);
  }
#pragma unroll
  for (int r = 0; r < 8; ++r) {
    const int m = m0 + r + (half << 3);
    const float sc = scale ? scale[m] : 1.f;
    const float bi = bias  ? bias[m]  : 0.f;
    const size_t rowo = ((size_t)batch * outStride + outOff + m) * NPIX + nbase + l16;
    float v0 = c0[r] * sc + bi;
    float v1 = c1[r] * sc + bi;
    float v2 = c2[r] * sc + bi;
    float v3 = c3[r] * sc + bi;
    if (addp) {
      v0 += addp[rowo +  0]; v1 += addp[rowo + 16];
      v2 += addp[rowo + 32]; v3 += addp[rowo + 48];
    }
    out[rowo +  0] = v0; out[rowo + 16] = v1;
    out[rowo + 32] = v2; out[rowo + 48] = v3;
  }
}

// ---------------------------------------------------------------------------
// Row softmax over n=4096 for the 16 k-channels of each batch (in place).
// ---------------------------------------------------------------------------
__global__ void softmax_rows(float* __restrict__ qkv) {
  const int b = blockIdx.x >> 4, k = blockIdx.x & 15;
  float* p = qkv + ((size_t)b * QKVC + 64 + k) * NPIX;
  __shared__ float red[256];
  const int t = threadIdx.x;
  float m = -3.4e38f;
  for (int i = t; i < NPIX; i += 256) m = fmaxf(m, p[i]);
  red[t] = m; __syncthreads();
  for (int s = 128; s > 0; s >>= 1) {
    if (t < s) red[t] = fmaxf(red[t], red[t + s]);
    __syncthreads();
  }
  m = red[0]; __syncthreads();
  float sum = 0.f;
  for (int i = t; i < NPIX; i += 256) { float e = __expf(p[i] - m); p[i] = e; sum += e; }
  red[t] = sum; __syncthreads();
  for (int s = 128; s > 0; s >>= 1) {
    if (t < s) red[t] += red[t + s];
    __syncthreads();
  }
  const float inv = 1.f / red[0];
  for (int i = t; i < NPIX; i += 256) p[i] *= inv;
}

// ---------------------------------------------------------------------------
// lam_c[b,k,v] = sum_n ksm[b,k,n] * v[b,v,n].  One block per (b,k).
// Each wave owns one v-row at a time; 32 lanes walk n contiguously
// (coalesced), then a 5-step shfl_xor wave reduction.
// ---------------------------------------------------------------------------
__global__ void lamc_kernel(const float* __restrict__ qkv, float* __restrict__ lamc) {
  const int b = blockIdx.x >> 4, k = blockIdx.x & 15;
  const float* ks = qkv + ((size_t)b * QKVC + 64 + k) * NPIX;
  const float* vb = qkv + ((size_t)b * QKVC + 80) * NPIX;
  const int lane = threadIdx.x & 31;
  const int wave = threadIdx.x >> 5;
  for (int vg = 0; vg < 8; ++vg) {            // 8 waves x 8 iters = 64 v rows
    const int v = vg * 8 + wave;
    const float* vp = vb + (size_t)v * NPIX;
    float acc = 0.f;
    for (int i = lane; i < NPIX; i += 32) acc += ks[i] * vp[i];
#pragma unroll
    for (int off = 16; off > 0; off >>= 1) acc += __shfl_xor(acc, off, 32);
    if (lane == 0) lamc[(size_t)b * 1024 + k * 64 + v] = acc;
  }
}

// ---------------------------------------------------------------------------
// Fused attention output:  out = x + y_c + y_p   (writes x_attn to d_out)
// Tile = 2x4 pixels per block, 256 threads = 8 px * 32 v-lanes.
// y_p via s-trick: s[px,h,tap] = sum_k q[px,h*16+k]*emb[k,tap] (LDS),
// then y_p[h,v] = sum_tap s * v(halo).  v halo staged with
// GLOBAL_LOAD_ASYNC_TO_LDS_B32 (ASYNCcnt) in two 32-ch halves (~51 KB LDS).
// ---------------------------------------------------------------------------
__global__ void attn_combine(const float* __restrict__ x, const float* __restrict__ qkv,
                             const float* __restrict__ lamc, const float* __restrict__ embw,
                             float* __restrict__ out) {
  const int b   = blockIdx.z;
  const int py0 = blockIdx.y * 2;
  const int px0 = blockIdx.x * 4;
  __shared__ float s_emb[16 * 121];       // 7.7 KB
  __shared__ float s_lamc[16 * 64];       // 4 KB
  __shared__ float s_q[8][64];            // 2 KB   (px, h*16+k)
  __shared__ float s_s[8][4][121];        // 15.5 KB
  __shared__ float s_v[12 * 14][32];      // 21.5 KB ([row*14+col][ch]) conflict-free reads
  const int t = threadIdx.x;

  for (int i = t; i < 16 * 121; i += 256) s_emb[i] = embw[i];
  for (int i = t; i < 16 * 64;  i += 256) s_lamc[i] = lamc[(size_t)b * 1024 + i];
  {
    const float* qb = qkv + (size_t)b * QKVC * NPIX;
    for (int i = t; i < 8 * 64; i += 256) {
      const int px = i >> 6, ch = i & 63;
      const int py = py0 + (px >> 2), pxx = px0 + (px & 3);
      s_q[px][ch] = qb[(size_t)ch * NPIX + py * 64 + pxx];
    }
  }
  __syncthreads();

  // s[px][h][tap] = sum_k q * emb
  for (int i = t; i < 8 * 4 * 121; i += 256) {
    const int px = i / 484, rem = i - px * 484;
    const int h = rem / 121, tap = rem - h * 121;
    float acc = 0.f;
#pragma unroll
    for (int k = 0; k < 16; ++k) acc += s_q[px][h * 16 + k] * s_emb[k * 121 + tap];
    s_s[px][h][tap] = acc;
  }

  const int px    = t >> 5;      // 0..7
  const int vlane = t & 31;
  const int prow  = px >> 2, pcol = px & 3;
  const int gy = py0 + prow, gx = px0 + pcol;
  const float* vbase = qkv + ((size_t)b * QKVC + 80) * NPIX;

  for (int halfc = 0; halfc < 2; ++halfc) {
    __syncthreads();   // previous-iteration consumers done before restaging
    // stage v halo tile: 32 ch x 12 rows x 14 cols; interior lanes use the
    // async Global->LDS DMA path, border lanes zero-fill via ds_store.
    for (int i = t; i < 32 * 12 * 14; i += 256) {
      const int ch = i / 168, rem = i - ch * 168;
      const int r = rem / 14, cc = rem - r * 14;
      const int yy = py0 - 5 + r, xx = px0 - 5 + cc;
      float* ldsp = &s_v[r * 14 + cc][ch];
      if (yy >= 0 && yy < 64 && xx >= 0 && xx < 64) {
        const float* gptr = vbase + (size_t)(halfc * 32 + ch) * NPIX + yy * 64 + xx;
        const unsigned ldsoff = (unsigned)(uintptr_t)ldsp;
        asm volatile("global_load_async_to_lds_b32 %0, %1, off"
                     :: "v"(ldsoff), "v"(gptr) : "memory");
      } else {
        *ldsp = 0.f;
      }
    }
    wait_asynccnt0();
    __syncthreads();

    const int vch = halfc * 32 + vlane;
    float a0 = 0.f, a1 = 0.f, a2 = 0.f, a3 = 0.f;
    // content lambda: y_c[h,v] = sum_k q[h,k] * lam_c[k,v]
#pragma unroll
    for (int k = 0; k < 16; ++k) {
      const float lc = s_lamc[k * 64 + vch];
      a0 += s_q[px][k]      * lc;
      a1 += s_q[px][16 + k] * lc;
      a2 += s_q[px][32 + k] * lc;
      a3 += s_q[px][48 + k] * lc;
    }
    // positional lambda: y_p[h,v] = sum_tap s[h,tap] * v[v, pix+tap-5]
    for (int dy = 0; dy < 11; ++dy) {
#pragma unroll
      for (int dx = 0; dx < 11; ++dx) {
        const float vv = s_v[(prow + dy) * 14 + (pcol + dx)][vlane];
        const int tap = dy * 11 + dx;
        a0 += s_s[px][0][tap] * vv;
        a1 += s_s[px][1][tap] * vv;
        a2 += s_s[px][2][tap] * vv;
        a3 += s_s[px][3][tap] * vv;
      }
    }
    const size_t pix = (size_t)b * 256 * NPIX + gy * 64 + gx;
    out[pix + (size_t)(0 * 64 + vch) * NPIX] = x[pix + (size_t)(0 * 64 + vch) * NPIX] + a0;
    out[pix + (size_t)(1 * 64 + vch) * NPIX] = x[pix + (size_t)(1 * 64 + vch) * NPIX] + a1;
    out[pix + (size_t)(2 * 64 + vch) * NPIX] = x[pix + (size_t)(2 * 64 + vch) * NPIX] + a2;
    out[pix + (size_t)(3 * 64 + vch) * NPIX] = x[pix + (size_t)(3 * 64 + vch) * NPIX] + a3;
  }
}

// ---------------------------------------------------------------------------
// Grouped 3x3 conv (16 groups, 4-in/4-out) + bias + exact GELU.
// Block = one (batch, group, 16x16 pixel tile); 4ch 18x18 halo in LDS.
// ---------------------------------------------------------------------------
__global__ void dwconv_gelu(const float* __restrict__ h1, const float* __restrict__ dw,
                            const float* __restrict__ bias, float* __restrict__ h2) {
  const int b = blockIdx.z;
  const int g = blockIdx.y;
  const int ty0 = (blockIdx.x >> 2) * 16, tx0 = (blockIdx.x & 3) * 16;
  __shared__ float tile[4][18][18];
  __shared__ float wts[144];               // [o*36 + i*9 + dy*3 + dx]
  const int t = threadIdx.x;
  for (int i = t; i < 144; i += 256) wts[i] = dw[(size_t)g * 144 + i];
  for (int i = t; i < 4 * 18 * 18; i += 256) {
    const int ch = i / 324, rem = i - ch * 324;
    const int r = rem / 18, cc = rem - r * 18;
    const int yy = ty0 - 1 + r, xx = tx0 - 1 + cc;
    float val = 0.f;
    if (yy >= 0 && yy < 64 && xx >= 0 && xx < 64)
      val = h1[((size_t)b * 64 + g * 4 + ch) * NPIX + yy * 64 + xx];
    tile[ch][r][cc] = val;
  }
  __syncthreads();
  const int ly = t >> 4, lx = t & 15;
#pragma unroll
  for (int o = 0; o < 4; ++o) {
    float acc = bias[g * 4 + o];
#pragma unroll
    for (int i = 0; i < 4; ++i)
#pragma unroll
      for (int dy = 0; dy < 3; ++dy)
#pragma unroll
        for (int dx = 0; dx < 3; ++dx)
          acc += wts[o * 36 + i * 9 + dy * 3 + dx] * tile[i][ly + dy][lx + dx];
    const float gel = 0.5f * acc * (1.f + erff(acc * 0.70710678118654752f));
    h2[((size_t)b * 64 + g * 4 + o) * NPIX + (ty0 + ly) * 64 + tx0 + lx] = gel;
  }
}

// ---------------------------------------------------------------------------
extern "C" void kernel_launch(void* const* d_in, const int* in_sizes, int n_in,
                              void* d_out, int out_size, void* d_ws, size_t ws_size,
                              hipStream_t stream) {
  (void)in_sizes; (void)n_in; (void)out_size; (void)ws_size;
  const float* x     = (const float*)d_in[0];
  const float* wq    = (const float*)d_in[1];
  const float* bnq_g = (const float*)d_in[2];
  const float* bnq_b = (const float*)d_in[3];
  const float* bnq_m = (const float*)d_in[4];
  const float* bnq_v = (const float*)d_in[5];
  const float* wk    = (const float*)d_in[6];
  const float* wv    = (const float*)d_in[7];
  const float* bnv_g = (const float*)d_in[8];
  const float* bnv_b = (const float*)d_in[9];
  const float* bnv_m = (const float*)d_in[10];
  const float* bnv_v = (const float*)d_in[11];
  const float* embw  = (const float*)d_in[12];
  const float* fc1_w = (const float*)d_in[13];
  const float* fc1_b = (const float*)d_in[14];
  const float* dw_w  = (const float*)d_in[15];
  const float* dw_b  = (const float*)d_in[16];
  const float* fc2_w = (const float*)d_in[17];
  const float* fc2_b = (const float*)d_in[18];
  float* out = (float*)d_out;

  float* ws   = (float*)d_ws;
  float* qkv  = ws + OFF_QKV;
  float* lamc = ws + OFF_LAMC;
  float* sbS  = ws + OFF_SBS;
  float* sbB  = ws + OFF_SBB;
  float* h1   = ws + OFF_H1;
  float* h2   = ws + OFF_H2;

  prep_scalebias<<<1, 256, 0, stream>>>(bnq_g, bnq_b, bnq_m, bnq_v,
                                        bnv_g, bnv_b, bnv_m, bnv_v, sbS, sbB);

  // qkv GEMMs (per-batch [M,256] @ [256,4096]), BN folded into epilogue
  gemm_wmma_f32<<<dim3(8, 4, BATCH),  256, 0, stream>>>(wq, x, sbS,      sbB,      nullptr, qkv, 64, 256, QKVC, 0);
  gemm_wmma_f32<<<dim3(8, 1, BATCH),  256, 0, stream>>>(wk, x, nullptr,  nullptr,  nullptr, qkv, 16, 256, QKVC, 64);
  gemm_wmma_f32<<<dim3(8, 4, BATCH),  256, 0, stream>>>(wv, x, sbS + 80, sbB + 80, nullptr, qkv, 64, 256, QKVC, 80);

  softmax_rows<<<BATCH * 16, 256, 0, stream>>>(qkv);
  lamc_kernel <<<BATCH * 16, 256, 0, stream>>>(qkv, lamc);

  // x_attn = x + y_c + y_p  -> d_out
  attn_combine<<<dim3(16, 32, BATCH), 256, 0, stream>>>(x, qkv, lamc, embw, out);

  // MLP: fc1 -> grouped dwconv + GELU -> fc2 + residual
  gemm_wmma_f32<<<dim3(8, 4, BATCH),  256, 0, stream>>>(fc1_w, out, nullptr, fc1_b, nullptr, h1, 64, 256, 64, 0);
  dwconv_gelu  <<<dim3(16, 16, BATCH), 256, 0, stream>>>(h1, dw_w, dw_b, h2);
  gemm_wmma_f32<<<dim3(8, 16, BATCH), 256, 0, stream>>>(fc2_w, h2, nullptr, fc2_b, out, out, 256, 64, 256, 0);
}